// CombinedLoss_29197187678599
// MI455X (gfx1250) — compile-verified
//
#include <hip/hip_runtime.h>
#include <hip/hip_bf16.h>

typedef __attribute__((ext_vector_type(2))) float v2f;
typedef __attribute__((ext_vector_type(8))) float v8f;

// Constants from the reference
#define GRID_N      256
#define INV255      (1.0f / 255.0f)
#define ELL_R2      0.09f               // ELL_RADIUS^2
#define GAU_R2      0.04f               // GAU_RADIUS^2
#define GAU_COEF    50.0f               // 1 / (2 * (SIGMA/SHARP)^2) = 1/0.02
#define ELL_CX      22.2222222f         // 0.5 / SIG_MAJ^2 = 0.5/0.0225
#define ELL_CY      200.0f              // 0.5 / SIG_MIN^2 = 0.5/0.0025
#define NLM         13

// Zero the padding tail of the workspace so the WMMA reduction can load
// unconditionally (no EXEC-masked loads in the hot loop).
__global__ void zero_pad_kernel(float* __restrict__ ws, int from, int to)
{
    const int i = from + blockIdx.x * blockDim.x + threadIdx.x;
    if (i < to) ws[i] = 0.0f;
}

// One workgroup (8 wave32s) per batch. Computes the four windowed reductions
// plus the cheap per-batch regression/BCE terms; writes one float per batch.
__global__ __launch_bounds__(256) void
per_batch_kernel(const float* __restrict__ predL, const float* __restrict__ tgtL,
                 const float* __restrict__ predV, const float* __restrict__ tgtV,
                 float* __restrict__ ws, int B)
{
    const int b   = blockIdx.x;
    const int tid = threadIdx.x;

    // Landmark 0 of batch b (uniform across block -> scalar loads)
    const float btx = tgtL[b * NLM * 2 + 0];
    const float bty = tgtL[b * NLM * 2 + 1];
    const float bpx = predL[b * NLM * 2 + 0];
    const float bpy = predL[b * NLM * 2 + 1];

    // Bounding-box window: outside |dx|>0.3 or |dy|>0.3 both weights are 0.
    // Expand by 1 px to be safe against float rounding at the boundary.
    int x0 = (int)ceilf ((btx - 0.3f) * 255.0f) - 1;
    int x1 = (int)floorf((btx + 0.3f) * 255.0f) + 1;
    int y0 = (int)ceilf ((bty - 0.3f) * 255.0f) - 1;
    int y1 = (int)floorf((bty + 0.3f) * 255.0f) + 1;
    x0 = max(x0, 0); y0 = max(y0, 0);
    x1 = min(x1, GRID_N - 1); y1 = min(y1, GRID_N - 1);

    float s_gwdp = 0.0f, s_gw = 0.0f, s_ewdp = 0.0f, s_ew = 0.0f;

    const int tx  = tid & 31;   // lane within wave32
    const int tyw = tid >> 5;   // wave index (0..7)

    for (int y = y0 + tyw; y <= y1; y += 8) {
        const float yg   = (float)y * INV255;
        const float dyt  = yg - bty;
        const float dyt2 = dyt * dyt;
        const float dyp  = yg - bpy;
        const float dyp2 = dyp * dyp;
        const float eyt  = dyt2 * ELL_CY;
        for (int x = x0 + tx; x <= x1; x += 32) {
            const float xg   = (float)x * INV255;
            const float dxt  = xg - btx;
            const float dxt2 = dxt * dxt;
            const float d2t  = dxt2 + dyt2;
            const float dxp  = xg - bpx;
            const float dp   = sqrtf(dxp * dxp + dyp2);   // only sqrt needed
            // branchless masks keep EXEC full; v_exp_f32 runs for all lanes
            const float ew = (d2t <= ELL_R2) ? __expf(-(dxt2 * ELL_CX + eyt)) : 0.0f;
            const float gw = (d2t <= GAU_R2) ? __expf(-GAU_COEF * d2t)        : 0.0f;
            s_ew   += ew;
            s_ewdp += ew * dp;
            s_gw   += gw;
            s_gwdp += gw * dp;
        }
    }

    // wave32 tree reduction (width 32 -- CDNA5 is wave32-only)
    for (int off = 16; off > 0; off >>= 1) {
        s_gwdp += __shfl_down(s_gwdp, off, 32);
        s_gw   += __shfl_down(s_gw,   off, 32);
        s_ewdp += __shfl_down(s_ewdp, off, 32);
        s_ew   += __shfl_down(s_ew,   off, 32);
    }

    __shared__ float red[8][4];
    if (tx == 0) {
        red[tyw][0] = s_gwdp; red[tyw][1] = s_gw;
        red[tyw][2] = s_ewdp; red[tyw][3] = s_ew;
    }
    __syncthreads();

    if (tid == 0) {
        float gwdp = 0.0f, gw = 0.0f, ewdp = 0.0f, ew = 0.0f;
        for (int w = 0; w < 8; ++w) {
            gwdp += red[w][0]; gw += red[w][1];
            ewdp += red[w][2]; ew += red[w][3];
        }
        const float g_per = gwdp / (gw + 1e-8f);
        const float e_per = ewdp / (ew + 1e-8f);

        const float visT    = tgtV[b * NLM];
        const float visible = (visT >= 0.5f) ? 1.0f : 0.0f;
        const float fB      = (float)B;

        // ELL_W = GAU_W = 1.0
        const float lm = (g_per + e_per) * visible / (fB + 1e-8f);

        // smooth-L1 over landmark-0 coords, mean over B*2 elements, REG_W=0.3
        const float adx = fabsf(bpx - btx);
        const float ady = fabsf(bpy - bty);
        const float hx  = (adx < 1.0f) ? 0.5f * adx * adx : adx - 0.5f;
        const float hy  = (ady < 1.0f) ? 0.5f * ady * ady : ady - 0.5f;
        const float reg = (hx + hy) / (2.0f * fB);

        // BCE on visibility, mean over B elements, VIS_W=0.01
        float p = predV[b * NLM];
        p = fminf(fmaxf(p, 1e-7f), 1.0f - 1e-7f);
        const float bce = -(visT * __logf(p) + (1.0f - visT) * __logf(1.0f - p)) / fB;

        ws[b] = lm + 0.3f * reg + 0.01f * bce;
    }
}

// Deterministic final reduction of the (zero-padded) per-batch floats using
// V_WMMA_F32_16X16X4_F32: D = A(16x4) * ones(4x16) + C accumulates row-sums
// of 64 values per issue. One wave32, fully uniform flow -> EXEC all ones,
// loads are unconditional global_load_b64 (each lane's two A-values are
// contiguous; the A-element->row mapping is irrelevant for a total sum).
__global__ __launch_bounds__(32) void
reduce_wmma_kernel(const float* __restrict__ ws, float* __restrict__ out, int iters)
{
    const int lane = threadIdx.x;

    v8f c = {};                               // C/D accumulator: 16x16 f32
    v2f ones; ones.x = 1.0f; ones.y = 1.0f;   // B = all-ones 4x16

    const float* p64 = ws + 2 * lane;
    for (int k = 0; k < iters; ++k) {
        const v2f a = *(const v2f*)(p64 + k * 64);   // one b64 load per lane
        // args: (neg_a, A, neg_b, B, c_mod, C, reuse_a, reuse_b)
        c = __builtin_amdgcn_wmma_f32_16x16x4_f32(false, a, false, ones,
                                                  (short)0, c, false, false);
    }

    // Every column of D holds the 16 row-sums. Column 0 lives in lane 0
    // (M=0..7 across the 8 accumulator VGPRs) and lane 16 (M=8..15).
    float p = c[0] + c[1] + c[2] + c[3] + c[4] + c[5] + c[6] + c[7];
    const float total = __shfl(p, 0, 32) + __shfl(p, 16, 32);
    if (lane == 0) out[0] = total;
}

extern "C" void kernel_launch(void* const* d_in, const int* in_sizes, int n_in,
                              void* d_out, int out_size, void* d_ws, size_t ws_size,
                              hipStream_t stream)
{
    const float* predL = (const float*)d_in[0];  // (B, 13, 2)
    const float* tgtL  = (const float*)d_in[1];  // (B, 13, 2)
    const float* predV = (const float*)d_in[2];  // (B, 13)
    const float* tgtV  = (const float*)d_in[3];  // (B, 13)

    const int B    = in_sizes[2] / NLM;          // 1024 for the reference setup
    const int Bpad = (B + 63) & ~63;             // multiple of 64 for WMMA tiles
    float* ws = (float*)d_ws;                    // Bpad floats of scratch

    if (Bpad > B) {  // deterministic: depends only on input sizes
        zero_pad_kernel<<<(Bpad - B + 63) / 64, 64, 0, stream>>>(ws, B, Bpad);
    }
    per_batch_kernel<<<B, 256, 0, stream>>>(predL, tgtL, predV, tgtV, ws, B);
    reduce_wmma_kernel<<<1, 32, 0, stream>>>(ws, (float*)d_out, Bpad / 64);
}